// DruckerPragerPlasticity_25323127177346
// MI455X (gfx1250) — compile-verified
//
#include <hip/hip_runtime.h>
#include <math.h>

#define TPB 256

// ---------------- fast math helpers ----------------
__device__ __forceinline__ float rsqrt_fast(float x) {
#if defined(__HIP_DEVICE_COMPILE__) && __has_builtin(__builtin_amdgcn_rsqf)
  return __builtin_amdgcn_rsqf(x);   // v_rsq_f32
#else
  return 1.0f / sqrtf(x);
#endif
}
__device__ __forceinline__ float rcp_fast(float x) {
#if defined(__HIP_DEVICE_COMPILE__) && __has_builtin(__builtin_amdgcn_rcpf)
  return __builtin_amdgcn_rcpf(x);   // v_rcp_f32
#else
  return 1.0f / x;
#endif
}

// ---------------- CDNA5 async global<->LDS ----------------
typedef int v4i __attribute__((vector_size(16)));
#define GPV(p) ((__attribute__((address_space(1))) v4i*)(p))   // global int4*
#define LPV(p) ((__attribute__((address_space(3))) v4i*)(p))   // LDS int4*

__device__ __forceinline__ void async_g2l_b128(void* lds_dst, const void* gsrc) {
#if defined(__HIP_DEVICE_COMPILE__)
#if __has_builtin(__builtin_amdgcn_global_load_async_to_lds_b128)
  __builtin_amdgcn_global_load_async_to_lds_b128(GPV((void*)gsrc), LPV(lds_dst), 0, 0);
#else
  unsigned l = (unsigned)(unsigned long long)LPV(lds_dst);
  asm volatile("global_load_async_to_lds_b128 %0, %1, off"
               :: "v"(l), "v"(gsrc) : "memory");
#endif
#else
  (void)lds_dst; (void)gsrc;
#endif
}

__device__ __forceinline__ void async_l2g_b128(void* gdst, void* lds_src) {
#if defined(__HIP_DEVICE_COMPILE__)
#if __has_builtin(__builtin_amdgcn_global_store_async_from_lds_b128)
  __builtin_amdgcn_global_store_async_from_lds_b128(GPV(gdst), LPV(lds_src), 0, 0);
#else
  unsigned l = (unsigned)(unsigned long long)LPV(lds_src);
  asm volatile("global_store_async_from_lds_b128 %0, %1, off"
               :: "v"(gdst), "v"(l) : "memory");
#endif
#else
  (void)gdst; (void)lds_src;
#endif
}

__device__ __forceinline__ void wait_async0() {
#if defined(__HIP_DEVICE_COMPILE__)
#if __has_builtin(__builtin_amdgcn_s_wait_asynccnt)
  __builtin_amdgcn_s_wait_asynccnt(0);
#else
  asm volatile("s_wait_asynccnt 0x0" ::: "memory");
#endif
#endif
}

// ---------------- branch-free 3x3 SVD pieces ----------------
__device__ __forceinline__ void cswap(bool c, float &x, float &y) {
  float t = x; x = c ? y : x; y = c ? t : y;
}
__device__ __forceinline__ void cns(bool c, float &x, float &y) { // cond-neg-swap
  float t = -x; x = c ? y : x; y = c ? t : y;
}

// One Jacobi conjugation on symmetric S (lower tri s11,s21,s22,s31,s32,s33),
// accumulating rotation into quaternion q; cyclically permutes S for next axis.
template<int X, int Y, int Z>
__device__ __forceinline__ void jacobi_conj(float &s11, float &s21, float &s22,
                                            float &s31, float &s32, float &s33,
                                            float q[4]) {
  float ch = 2.0f * (s11 - s22);
  float sh = s21;
  bool ok = (5.8284271247461903f * sh * sh) < (ch * ch);
  float w = rsqrt_fast(ch * ch + sh * sh);
  float chn = ok ? w * ch : 0.9238795325112867f;   // cos(pi/8)
  float shn = ok ? w * sh : 0.3826834323650898f;   // sin(pi/8)
  float a = chn * chn - shn * shn;
  float b = 2.0f * shn * chn;

  float _s11=s11,_s21=s21,_s22=s22,_s31=s31,_s32=s32,_s33=s33;
  s11 =  a*( a*_s11 + b*_s21) + b*( a*_s21 + b*_s22);
  s21 =  a*(-b*_s11 + a*_s21) + b*(-b*_s21 + a*_s22);
  s22 = -b*(-b*_s11 + a*_s21) + a*(-b*_s21 + a*_s22);
  s31 =  a*_s31 + b*_s32;
  s32 = -b*_s31 + a*_s32;
  s33 = _s33;

  float t0 = q[0]*shn, t1 = q[1]*shn, t2 = q[2]*shn;
  float tmp[3] = {t0, t1, t2};
  float shw = shn * q[3];
  q[0]*=chn; q[1]*=chn; q[2]*=chn; q[3]*=chn;
  q[Z] += shw;
  q[3] -= tmp[Z];
  q[X] += tmp[Y];
  q[Y] -= tmp[X];

  float n11=s22, n21=s32, n22=s33, n31=s21, n32=s31, n33=s11;
  s11=n11; s21=n21; s22=n22; s31=n31; s32=n32; s33=n33;
}

__device__ __forceinline__ void qr_givens(float a1, float a2, float &ch, float &sh) {
  float rho = sqrtf(a1*a1 + a2*a2);
  sh = (rho > 1e-12f) ? a2 : 0.0f;
  ch = fabsf(a1) + fmaxf(rho, 1e-12f);
  bool neg = a1 < 0.0f;
  cswap(neg, sh, ch);
  float w = rsqrt_fast(ch*ch + sh*sh);
  ch *= w; sh *= w;
}

// ---------------- main kernel ----------------
extern "C" __global__ __launch_bounds__(TPB)
void dp_plasticity_kernel(const float* __restrict__ Fg,
                          const float* __restrict__ ylog_p,
                          const float* __restrict__ nu_p,
                          const float* __restrict__ fric_p,
                          const float* __restrict__ coh_p,
                          float* __restrict__ outg,
                          int B)
{
  __shared__ __align__(16) float ldsIn[TPB * 9];   // 9216 B = 576 x 16B
  __shared__ __align__(16) float ldsOut[TPB * 9];

  const int tid  = threadIdx.x;
  const int base = blockIdx.x * TPB;
  const bool full = (base + TPB) <= B;

  const float* gin  = Fg   + (size_t)base * 9;
  float*       gout = outg + (size_t)base * 9;

  // ---- kick off input staging to LDS (async b128, perfectly coalesced) ----
  if (full) {
#pragma unroll
    for (int k = 0; k < 3; ++k) {
      int c = tid + k * TPB;                 // 0..575 chunks of 16B
      if (c < 576) async_g2l_b128(&ldsIn[c * 4], gin + c * 4);
    }
  } else {
    int nf = (B - base) * 9;
    for (int i = tid; i < nf; i += TPB) ldsIn[i] = gin[i];
  }

  // ---- uniform material constants: overlap with async-load latency ----
  float E    = __expf(ylog_p[0]);
  float nu   = nu_p[0];
  float sphi = __sinf(fric_p[0] * 0.017453292519943295f);
  float alpha = 0.8164965809277260f * 2.0f * sphi / (3.0f - sphi); // sqrt(2/3)*2 sin/(3-sin)
  float mu = E / (2.0f * (1.0f + nu));
  float la = E * nu / ((1.0f + nu) * (1.0f - 2.0f * nu));
  float ratio = (3.0f * la + 2.0f * mu) / (2.0f * mu);
  float coh = coh_p[0];

  if (full) wait_async0();
  __syncthreads();

  if (base + tid < B) {
    const float* m = &ldsIn[tid * 9];        // stride 9 dwords: bank-conflict free
    float f00=m[0], f01=m[1], f02=m[2];
    float f10=m[3], f11=m[4], f12=m[5];
    float f20=m[6], f21=m[7], f22=m[8];

    // ---- S = F^T F ----
    float s11 = f00*f00 + f10*f10 + f20*f20;
    float s21 = f00*f01 + f10*f11 + f20*f21;
    float s22 = f01*f01 + f11*f11 + f21*f21;
    float s31 = f00*f02 + f10*f12 + f20*f22;
    float s32 = f01*f02 + f11*f12 + f21*f22;
    float s33 = f02*f02 + f12*f12 + f22*f22;

    // ---- Jacobi eigenanalysis (4 cyclic sweeps, quaternion accumulation) ----
    float q[4] = {0.f, 0.f, 0.f, 1.f};
#pragma unroll
    for (int it = 0; it < 4; ++it) {
      jacobi_conj<0,1,2>(s11,s21,s22,s31,s32,s33,q);
      jacobi_conj<1,2,0>(s11,s21,s22,s31,s32,s33,q);
      jacobi_conj<2,0,1>(s11,s21,s22,s31,s32,s33,q);
    }
    float qn = rsqrt_fast(q[0]*q[0]+q[1]*q[1]+q[2]*q[2]+q[3]*q[3]);
    float qx=q[0]*qn, qy=q[1]*qn, qz=q[2]*qn, qw=q[3]*qn;

    float v00 = 1.f - 2.f*(qy*qy + qz*qz);
    float v01 = 2.f*(qx*qy - qw*qz);
    float v02 = 2.f*(qx*qz + qw*qy);
    float v10 = 2.f*(qx*qy + qw*qz);
    float v11 = 1.f - 2.f*(qx*qx + qz*qz);
    float v12 = 2.f*(qy*qz - qw*qx);
    float v20 = 2.f*(qx*qz - qw*qy);
    float v21 = 2.f*(qy*qz + qw*qx);
    float v22 = 1.f - 2.f*(qx*qx + qy*qy);

    // ---- B = F * V ----
    float b00 = f00*v00 + f01*v10 + f02*v20;
    float b01 = f00*v01 + f01*v11 + f02*v21;
    float b02 = f00*v02 + f01*v12 + f02*v22;
    float b10 = f10*v00 + f11*v10 + f12*v20;
    float b11 = f10*v01 + f11*v11 + f12*v21;
    float b12 = f10*v02 + f11*v12 + f12*v22;
    float b20 = f20*v00 + f21*v10 + f22*v20;
    float b21 = f20*v01 + f21*v11 + f22*v21;
    float b22 = f20*v02 + f21*v12 + f22*v22;

    // ---- sort columns by magnitude (branch-free cond-neg-swaps) ----
    float rho1 = b00*b00 + b10*b10 + b20*b20;
    float rho2 = b01*b01 + b11*b11 + b21*b21;
    float rho3 = b02*b02 + b12*b12 + b22*b22;
    bool c;
    c = rho1 < rho2;
    cns(c,b00,b01); cns(c,b10,b11); cns(c,b20,b21);
    cns(c,v00,v01); cns(c,v10,v11); cns(c,v20,v21);
    cswap(c, rho1, rho2);
    c = rho1 < rho3;
    cns(c,b00,b02); cns(c,b10,b12); cns(c,b20,b22);
    cns(c,v00,v02); cns(c,v10,v12); cns(c,v20,v22);
    cswap(c, rho1, rho3);
    c = rho2 < rho3;
    cns(c,b01,b02); cns(c,b11,b12); cns(c,b21,b22);
    cns(c,v01,v02); cns(c,v11,v12); cns(c,v21,v22);
    cswap(c, rho2, rho3);

    // ---- QR of B via 3 Givens rotations: B = U R, sigma = diag(R) ----
    float ch1, sh1; qr_givens(b00, b10, ch1, sh1);
    float a1 = 1.f - 2.f*sh1*sh1, g1 = 2.f*ch1*sh1;
    float r00 =  a1*b00 + g1*b10, r01 =  a1*b01 + g1*b11, r02 =  a1*b02 + g1*b12;
    float r11 = -g1*b01 + a1*b11, r12 = -g1*b02 + a1*b12;
    float r20 = b20, r21 = b21, r22 = b22;

    float ch2, sh2; qr_givens(r00, r20, ch2, sh2);
    float a2 = 1.f - 2.f*sh2*sh2, g2 = 2.f*ch2*sh2;
    float t00 =  a2*r00 + g2*r20;
    float t11 = r11, t12 = r12;
    float t21 = -g2*r01 + a2*r21, t22 = -g2*r02 + a2*r22;

    float ch3, sh3; qr_givens(t11, t21, ch3, sh3);
    float a3 = 1.f - 2.f*sh3*sh3, g3 = 2.f*ch3*sh3;

    float sg1 = t00;
    float sg2 =  a3*t11 + g3*t21;
    float sg3 = -g3*t12 + a3*t22;

    // U = G1^T G2^T G3^T
    float u00 = a1*a2;
    float u01 = -a1*g2*g3 - g1*a3;
    float u02 = -a1*g2*a3 + g1*g3;
    float u10 = g1*a2;
    float u11 = -g1*g2*g3 + a1*a3;
    float u12 = -g1*g2*a3 - a1*g3;
    float u20 = g2;
    float u21 = a2*g3;
    float u22 = a2*a3;

    // SVD convention: non-negative singular values (flip U col 3 with sg3)
    float sgn = (sg3 < 0.f) ? -1.f : 1.f;
    sg3 *= sgn; u02 *= sgn; u12 *= sgn; u22 *= sgn;

    // ---- Drucker-Prager return mapping in log-strain space ----
    float e1 = __logf(fmaxf(sg1, 0.05f));
    float e2 = __logf(fmaxf(sg2, 0.05f));
    float e3 = __logf(fmaxf(sg3, 0.05f));
    float tr = e1 + e2 + e3;
    float third = tr * (1.0f / 3.0f);
    float h1 = e1 - third, h2 = e2 - third, h3 = e3 - third;
    float hn = fmaxf(sqrtf(h1*h1 + h2*h2 + h3*h3), 1e-10f);
    float shifted = tr - coh * 3.0f;
    float dg = hn + ratio * shifted * alpha;
    float sc = fmaxf(dg, 0.0f) * rcp_fast(hn);
    bool yield = shifted < 0.0f;
    float ec1 = yield ? (e1 - sc*h1) : coh;
    float ec2 = yield ? (e2 - sc*h2) : coh;
    float ec3 = yield ? (e3 - sc*h3) : coh;
    float d1 = __expf(ec1), d2 = __expf(ec2), d3 = __expf(ec3);

    // ---- F_corrected = U diag(d) V^T ----
    float* o = &ldsOut[tid * 9];
    float w0, w1, w2;
    w0 = u00*d1; w1 = u01*d2; w2 = u02*d3;
    o[0] = w0*v00 + w1*v01 + w2*v02;
    o[1] = w0*v10 + w1*v11 + w2*v12;
    o[2] = w0*v20 + w1*v21 + w2*v22;
    w0 = u10*d1; w1 = u11*d2; w2 = u12*d3;
    o[3] = w0*v00 + w1*v01 + w2*v02;
    o[4] = w0*v10 + w1*v11 + w2*v12;
    o[5] = w0*v20 + w1*v21 + w2*v22;
    w0 = u20*d1; w1 = u21*d2; w2 = u22*d3;
    o[6] = w0*v00 + w1*v01 + w2*v02;
    o[7] = w0*v10 + w1*v11 + w2*v12;
    o[8] = w0*v20 + w1*v21 + w2*v22;
  }

  __syncthreads();

  // ---- stage outputs back out (async b128 from LDS); S_ENDPGM waits idle ----
  if (full) {
#pragma unroll
    for (int k = 0; k < 3; ++k) {
      int c = tid + k * TPB;
      if (c < 576) async_l2g_b128(gout + c * 4, &ldsOut[c * 4]);
    }
  } else {
    int nf = (B - base) * 9;
    for (int i = tid; i < nf; i += TPB) gout[i] = ldsOut[i];
  }
}

// ---------------- host-side launcher ----------------
extern "C" void kernel_launch(void* const* d_in, const int* in_sizes, int n_in,
                              void* d_out, int out_size, void* d_ws, size_t ws_size,
                              hipStream_t stream) {
  (void)n_in; (void)out_size; (void)d_ws; (void)ws_size;
  const float* F  = (const float*)d_in[0];
  const float* yl = (const float*)d_in[1];
  const float* pr = (const float*)d_in[2];
  const float* fa = (const float*)d_in[3];
  const float* co = (const float*)d_in[4];
  float* out = (float*)d_out;
  int B = in_sizes[0] / 9;
  int grid = (B + TPB - 1) / TPB;
  dp_plasticity_kernel<<<dim3(grid), dim3(TPB), 0, stream>>>(F, yl, pr, fa, co, out, B);
}